// RAA_10110353015379
// MI455X (gfx1250) — compile-verified
//
#include <hip/hip_runtime.h>
#include <math.h>

#define NN 4096
#define KK 8

typedef __attribute__((ext_vector_type(2))) float v2f;
typedef __attribute__((ext_vector_type(8))) float v8f;

__device__ __forceinline__ float block_reduce_sum(float v, float* red) {
  int tid = threadIdx.x;
  red[tid] = v;
  __syncthreads();
  for (int s = 128; s > 0; s >>= 1) {
    if (tid < s) red[tid] += red[tid + s];
    __syncthreads();
  }
  float r = red[0];
  __syncthreads();
  return r;
}

__device__ __forceinline__ float block_reduce_max(float v, float* red) {
  int tid = threadIdx.x;
  red[tid] = v;
  __syncthreads();
  for (int s = 128; s > 0; s >>= 1) {
    if (tid < s) red[tid] = fmaxf(red[tid], red[tid + s]);
    __syncthreads();
  }
  float r = red[0];
  __syncthreads();
  return r;
}

// Kernel 1: W[k][k'] = (Zs @ Cs)[k,k'] ; single block, 256 threads.
__global__ void __launch_bounds__(256) k1_weights(const float* __restrict__ Z,
                                                  const float* __restrict__ C,
                                                  float* __restrict__ W) {
  __shared__ float red[256];
  int tid = threadIdx.x;

  // column-max of C (softmax over N per column k')
  float cmax[KK];
#pragma unroll
  for (int k = 0; k < KK; ++k) cmax[k] = -3.402823466e+38f;
  for (int n = tid; n < NN; n += 256) {
#pragma unroll
    for (int k = 0; k < KK; ++k) cmax[k] = fmaxf(cmax[k], C[n * KK + k]);
  }
#pragma unroll
  for (int k = 0; k < KK; ++k) cmax[k] = block_reduce_max(cmax[k], red);

  float denom[KK];
  float num[KK][KK];
#pragma unroll
  for (int k = 0; k < KK; ++k) {
    denom[k] = 0.f;
#pragma unroll
    for (int kp = 0; kp < KK; ++kp) num[k][kp] = 0.f;
  }

  for (int n = tid; n < NN; n += 256) {
    // Zs[:,n] = softmax over K of Z[:,n]
    float z[KK];
    float zm = -3.402823466e+38f;
#pragma unroll
    for (int k = 0; k < KK; ++k) { z[k] = Z[k * NN + n]; zm = fmaxf(zm, z[k]); }
    float zs = 0.f;
#pragma unroll
    for (int k = 0; k < KK; ++k) { z[k] = expf(z[k] - zm); zs += z[k]; }
    float inv = 1.f / zs;
#pragma unroll
    for (int k = 0; k < KK; ++k) z[k] *= inv;

    float e[KK];
#pragma unroll
    for (int kp = 0; kp < KK; ++kp) {
      e[kp] = expf(C[n * KK + kp] - cmax[kp]);
      denom[kp] += e[kp];
    }
#pragma unroll
    for (int k = 0; k < KK; ++k)
#pragma unroll
      for (int kp = 0; kp < KK; ++kp) num[k][kp] += z[k] * e[kp];
  }

  float dsum[KK];
#pragma unroll
  for (int kp = 0; kp < KK; ++kp) dsum[kp] = block_reduce_sum(denom[kp], red);
#pragma unroll
  for (int k = 0; k < KK; ++k)
#pragma unroll
    for (int kp = 0; kp < KK; ++kp) {
      float s = block_reduce_sum(num[k][kp], red);
      if (tid == 0) W[k * KK + kp] = s / dsum[kp];
    }
}

// Kernel 2: M[n,k] = sum_k' W[k,k'] * Zs[k',n]; plus P[n], Q[n] row terms.
__global__ void __launch_bounds__(256) k2_latent(const float* __restrict__ Z,
                                                 const float* __restrict__ Wg,
                                                 float* __restrict__ M,
                                                 float* __restrict__ P,
                                                 float* __restrict__ Q) {
  __shared__ float sW[KK * KK];
  int tid = threadIdx.x;
  if (tid < KK * KK) sW[tid] = Wg[tid];
  __syncthreads();
  int n = blockIdx.x * 256 + tid;
  if (n >= NN) return;

  float z[KK];
  float zm = -3.402823466e+38f;
#pragma unroll
  for (int k = 0; k < KK; ++k) { z[k] = Z[k * NN + n]; zm = fmaxf(zm, z[k]); }
  float zs = 0.f;
#pragma unroll
  for (int k = 0; k < KK; ++k) { z[k] = expf(z[k] - zm); zs += z[k]; }
  float inv = 1.f / zs;
#pragma unroll
  for (int k = 0; k < KK; ++k) z[k] *= inv;

  float nrm = 0.f, S = 0.f;
#pragma unroll
  for (int k = 0; k < KK; ++k) {
    float acc = 0.f;
#pragma unroll
    for (int kp = 0; kp < KK; ++kp) acc += sW[k * KK + kp] * z[kp];
    M[(size_t)n * KK + k] = acc;
    nrm += acc * acc;
    S += acc;
  }
  const float eps = 1e-6f;
  P[n] = nrm + 2.f * eps * S + (float)KK * eps * eps;  // row-i part (+ constant)
  Q[n] = nrm - 2.f * eps * S;                          // row-j part
}

// Kernel 3: per 16x16 tile — G = M·M^T via chained f32 WMMA, fused epilogue.
// 1024 blocks x 256 threads = 8192 waves; wave = (iTile, jChunk of 8 j-tiles).
__global__ void __launch_bounds__(256) k3_main(const float* __restrict__ A,
                                               const float* __restrict__ beta,
                                               const float* __restrict__ a_ptr,
                                               const float* __restrict__ M,
                                               const float* __restrict__ P,
                                               const float* __restrict__ Q,
                                               float* __restrict__ partials) {
  __shared__ float s_wsum[8];
  const float a = a_ptr[0];
  int tid = threadIdx.x;
  int lane = tid & 31;
  int waveInBlk = tid >> 5;
  int waveGlobal = blockIdx.x * 8 + waveInBlk;
  int iTile = waveGlobal >> 5;   // 0..255
  int jChunk = waveGlobal & 31;  // 0..31 (each = 8 j-tiles)
  int r = lane & 15;
  int h = lane >> 4;
  int i0 = iTile * 16;

  // WMMA A operand (16x4 f32): lane holds row i0+r, K elems {2h,2h+1} / {4+2h,4+2h+1}
  v2f a0 = *(const v2f*)(M + (size_t)(i0 + r) * KK + 2 * h);
  v2f a1 = *(const v2f*)(M + (size_t)(i0 + r) * KK + 4 + 2 * h);

  float Pi[8], Bi[8];
#pragma unroll
  for (int v = 0; v < 8; ++v) {
    int i = i0 + v + 8 * h;
    Pi[v] = P[i];
    Bi[v] = beta[i];
  }

  float acc = 0.f;
  for (int t = 0; t < 8; ++t) {
    int j0 = (jChunk * 8 + t) * 16;
    // B = M^T: identical per-lane load pattern on the j rows
    v2f b0 = *(const v2f*)(M + (size_t)(j0 + r) * KK + 2 * h);
    v2f b1 = *(const v2f*)(M + (size_t)(j0 + r) * KK + 4 + 2 * h);
    v8f c = {};
    c = __builtin_amdgcn_wmma_f32_16x16x4_f32(false, a0, false, b0, (short)0, c,
                                              false, false);
    c = __builtin_amdgcn_wmma_f32_16x16x4_f32(false, a1, false, b1, (short)0, c,
                                              false, false);
    int j = j0 + r;
    float Qj = Q[j];
    float Bj = beta[j];
#pragma unroll
    for (int v = 0; v < 8; ++v) {
      int i = i0 + v + 8 * h;
      float d2 = Pi[v] + Qj - 2.f * c[v];
      // raw v_sqrt_f32: d2 ~ O(1e-3), far from denormal range, ~1 ulp is fine
      float zd = __builtin_amdgcn_sqrtf(fmaxf(d2, 0.f));
      float th = Bi[v] + Bj - a * zd;
      // A streamed exactly once -> non-temporal
      float Aij = __builtin_nontemporal_load(A + (size_t)i * NN + j);
      float e = __expf(-fabsf(th));
      float sp = fmaxf(th, 0.f) + __logf(1.f + e);
      if (i != j) acc += th * Aij - sp;
    }
  }

  // wave32 reduction -> LDS -> per-block partial (deterministic)
#pragma unroll
  for (int off = 16; off > 0; off >>= 1) acc += __shfl_down(acc, off, 32);
  if (lane == 0) s_wsum[waveInBlk] = acc;
  __syncthreads();
  if (tid == 0) {
    float s = 0.f;
#pragma unroll
    for (int w = 0; w < 8; ++w) s += s_wsum[w];
    partials[blockIdx.x] = s;
  }
}

// Kernel 4: deterministic final sum of 1024 partials.
__global__ void __launch_bounds__(256) k4_finish(const float* __restrict__ partials,
                                                 float* __restrict__ out, int n) {
  __shared__ float red[256];
  float v = 0.f;
  for (int i = threadIdx.x; i < n; i += 256) v += partials[i];
  v = block_reduce_sum(v, red);
  if (threadIdx.x == 0) out[0] = v;
}

extern "C" void kernel_launch(void* const* d_in, const int* in_sizes, int n_in,
                              void* d_out, int out_size, void* d_ws, size_t ws_size,
                              hipStream_t stream) {
  const float* A    = (const float*)d_in[0];  // [N,N]
  const float* beta = (const float*)d_in[1];  // [N]
  const float* a    = (const float*)d_in[2];  // [1]
  const float* Z    = (const float*)d_in[3];  // [K,N]
  const float* C    = (const float*)d_in[4];  // [N,K]

  float* ws = (float*)d_ws;
  float* W        = ws;                  // 64
  float* M        = W + KK * KK;         // N*K = 32768
  float* P        = M + (size_t)NN * KK; // N
  float* Q        = P + NN;              // N
  float* partials = Q + NN;              // 1024

  k1_weights<<<1, 256, 0, stream>>>(Z, C, W);
  k2_latent<<<NN / 256, 256, 0, stream>>>(Z, W, M, P, Q);
  k3_main<<<1024, 256, 0, stream>>>(A, beta, a, M, P, Q, partials);
  k4_finish<<<1, 256, 0, stream>>>(partials, (float*)d_out, 1024);
}